// SemlinkLoss_32899449487485
// MI455X (gfx1250) — compile-verified
//
#include <hip/hip_runtime.h>
#include <stdint.h>

#define Bc 8
#define Tc 256
#define Vc 16
#define Kc 8
#define LSRL 66
#define LVN  40

typedef float v2f __attribute__((ext_vector_type(2)));
typedef float v8f __attribute__((ext_vector_type(8)));
typedef unsigned int u32x4 __attribute__((ext_vector_type(4)));
typedef int i32x4 __attribute__((ext_vector_type(4)));
typedef int i32x8 __attribute__((ext_vector_type(8)));

#if __has_builtin(__builtin_amdgcn_tensor_load_to_lds)
#define HAVE_TDM 1
#else
#define HAVE_TDM 0
#endif

#if HAVE_TDM
// Stage a 2D tile (tile_w columns of a tensor_w-wide, rows-tall f32 tensor)
// from global memory into LDS via the Tensor Data Mover.
__device__ __forceinline__ void tdm_load_2d(const void* gptr, uint32_t lds_off,
                                            uint32_t tensor_w, uint32_t tile_w,
                                            uint32_t rows) {
  uint64_t ga = (uint64_t)(uintptr_t)gptr;
  u32x4 g0;
  g0[0] = 1u;                                           // count=1, user descriptor
  g0[1] = lds_off;                                      // lds_addr (bytes)
  g0[2] = (uint32_t)(ga & 0xFFFFFFFFu);                 // global_addr[31:0]
  g0[3] = (uint32_t)((ga >> 32) & 0x01FFFFFFu)          // global_addr[56:32]
        | (2u << 30);                                   // type = 2 ("image")
  i32x8 g1;
  g1[0] = (int)(2u << 16);                              // data_size = 4 bytes; no multicast
  g1[1] = (int)((tensor_w & 0xFFFFu) << 16);            // tensor_dim0[15:0] @ bits 63:48
  g1[2] = (int)(((tensor_w >> 16) & 0xFFFFu)            // tensor_dim0[31:16]
        | ((rows & 0xFFFFu) << 16));                    // tensor_dim1[15:0]
  g1[3] = (int)(((rows >> 16) & 0xFFFFu)                // tensor_dim1[31:16]
        | ((tile_w & 0xFFFFu) << 16));                  // tile_dim0
  g1[4] = (int)(rows & 0xFFFFu);                        // tile_dim1; tile_dim2 = 0
  g1[5] = (int)tensor_w;                                // tensor_dim0_stride[31:0]
  g1[6] = 0;                                            // stride hi / dim1_stride (unused, 2D)
  g1[7] = 0;
  i32x4 z4 = {0, 0, 0, 0};
#if defined(__clang_major__) && (__clang_major__ >= 23)
  i32x8 z8 = {0, 0, 0, 0, 0, 0, 0, 0};
  __builtin_amdgcn_tensor_load_to_lds(g0, g1, z4, z4, z8, 0);
#else
  __builtin_amdgcn_tensor_load_to_lds(g0, g1, z4, z4, 0);
#endif
}
#endif

__global__ void semlink_main(const float* __restrict__ log_srl,
                             const float* __restrict__ log_vn,
                             const int* __restrict__ v_label,
                             const int* __restrict__ v_l,
                             const int* __restrict__ orig_l,
                             const int* __restrict__ semlink,
                             const int* __restrict__ semlink_l,
                             float* __restrict__ block_out) {
  __shared__ float wavesum[8];
  extern __shared__ float smem[];            // [Tc*LVN] srl-tile, then [Tc*LVN] vn-tile
  const int bv = blockIdx.x;
  const int b = bv / Vc;
  const int v = bv % Vc;
  const int tid = threadIdx.x;
  const int vl = v_l[b];                     // uniform across the block

  if (v < vl) {
    const int lbl = v_label[b * Vc + v];
    const int kl  = semlink_l[b * Vc + v];
    const int tl  = orig_l[b];

    float* s_srl = smem;                     // Tc x 40 (only cols [0,40) are ever indexed)
    float* s_vn  = smem + Tc * LVN;          // Tc x 40

    const float* gsrl = log_srl + (((size_t)b * Tc + lbl) * Tc) * LSRL;
    const float* gvn  = log_vn  + (((size_t)b * Tc + lbl) * Tc) * LVN;

#if HAVE_TDM
    if (tid < 32) {                          // wave 0 issues both DMA tiles
      tdm_load_2d(gsrl, (uint32_t)(uintptr_t)s_srl, LSRL, LVN, Tc);
      tdm_load_2d(gvn,  (uint32_t)(uintptr_t)s_vn,  LVN,  LVN, Tc);
      __builtin_amdgcn_s_wait_tensorcnt(0);
    }
#else
    for (int i = tid; i < Tc * LVN; i += 256) {
      int t = i / LVN, x = i - t * LVN;
      s_srl[i] = gsrl[t * LSRL + x];
      s_vn[i]  = gvn[i];
    }
#endif
    __syncthreads();

    // Gather indices: semlink[b, v, 0, k] = r, semlink[b, v, 1, k] = a.
    // All Kc slots hold valid indices in [0, LVN); only k < kl contribute.
    int r[Kc], a[Kc];
    const int sbase = ((b * Vc + v) * 2) * Kc;
#pragma unroll
    for (int k = 0; k < Kc; ++k) {
      r[k] = semlink[sbase + k];
      a[k] = semlink[sbase + Kc + k];
    }

    // One t per thread (orig_l >= Tc/2, blockDim == Tc). Fully unroll over k so
    // all 16 ds_loads issue before a single dscnt wait, then mask contributions.
    float partial = 0.f;
    if (tid < tl) {
      const float* prow = s_srl + tid * LVN;
      const float* qrow = s_vn  + tid * LVN;
      float g_r[Kc], g_a[Kc];
#pragma unroll
      for (int k = 0; k < Kc; ++k) {
        g_r[k] = prow[r[k]];
        g_a[k] = qrow[a[k]];
      }
#pragma unroll
      for (int k = 0; k < Kc; ++k) {
        float dterm = fabsf(g_r[k] - g_a[k]);
        partial += (k < kl) ? dterm : 0.f;
      }
    }

    // Wave-level reduction via V_WMMA_F32_16X16X4_F32:
    // A[m][0]=partial(lane m), A[m][2]=partial(lane m+16), A[.][1]=A[.][3]=0;
    // B = ones  =>  D[m][n] = partial_m + partial_{m+16}.
    // Lane n<16 holds D rows 0..7, lane n>=16 holds rows 8..15 across the 8 acc VGPRs.
    v2f av; av.x = partial; av.y = 0.f;
    v2f bo; bo.x = 1.f;     bo.y = 1.f;
    v8f c = {};
    v8f d = __builtin_amdgcn_wmma_f32_16x16x4_f32(false, av, false, bo,
                                                  (short)0, c, false, false);
    float s = d[0] + d[1] + d[2] + d[3] + d[4] + d[5] + d[6] + d[7];
    s += __shfl_xor(s, 16, 32);              // combine the two row halves -> full 32-lane sum

    if ((tid & 31) == 0) wavesum[tid >> 5] = s;
    __syncthreads();
    if (tid == 0) {
      float tot = 0.f;
#pragma unroll
      for (int w = 0; w < 8; ++w) tot += wavesum[w];
      block_out[bv] = tot;
    }
  } else {
    if (tid == 0) block_out[bv] = 0.f;       // masked-out proposition slot
  }
}

__global__ void semlink_finalize(const float* __restrict__ block_out,
                                 const int* __restrict__ v_l,
                                 float* __restrict__ out) {
  float loss = 0.f;
  for (int i = 0; i < Bc * Vc; ++i) loss += block_out[i];
  int np = 0;
  for (int i = 0; i < Bc; ++i) np += v_l[i];
  out[0] = loss / (float)np;
}

extern "C" void kernel_launch(void* const* d_in, const int* in_sizes, int n_in,
                              void* d_out, int out_size, void* d_ws, size_t ws_size,
                              hipStream_t stream) {
  (void)in_sizes; (void)n_in; (void)out_size; (void)ws_size;
  const float* log_srl   = (const float*)d_in[0];
  const float* log_vn    = (const float*)d_in[1];
  const int*   v_label   = (const int*)d_in[2];
  const int*   v_l       = (const int*)d_in[3];
  const int*   orig_l    = (const int*)d_in[4];
  const int*   semlink   = (const int*)d_in[5];
  const int*   semlink_l = (const int*)d_in[6];
  float* block_out = (float*)d_ws;           // 128 floats of scratch

  const size_t shmem = (size_t)(2 * Tc * LVN) * sizeof(float);  // 80 KB
  semlink_main<<<Bc * Vc, 256, shmem, stream>>>(log_srl, log_vn, v_label, v_l,
                                                orig_l, semlink, semlink_l,
                                                block_out);
  semlink_finalize<<<1, 1, 0, stream>>>(block_out, v_l, (float*)d_out);
}